// RelationalDense_8400956031462
// MI455X (gfx1250) — compile-verified
//
#include <hip/hip_runtime.h>

typedef __attribute__((ext_vector_type(2))) float v2f;
typedef __attribute__((ext_vector_type(4))) float v4f;
typedef __attribute__((ext_vector_type(8))) float v8f;

#define NROWS    65536
#define FDIM     64
#define UNITS    64
#define NREL     25
#define NBASE    8
#define MAXTILES 4121            // sum ceil(count_r/16) <= 4096 + 25
#define NPAD     (MAXTILES * 16)
#define WSZ      (FDIM * UNITS)  // 4096 floats per relation

// ---------------------------------------------------------------------------
// ws layout: [0 .. 25*4096) floats  : Wpack, WMMA-B-packed per-relation weights
//   Wpack[((r*16 + k)*2 + g)*128 + lane*4 + q] =
//       W_r[(4k + 2*(lane>>4) + (q&1))*64 + (2g + (q>>1))*16 + (lane&15)]
// then int region ibase = (int*)ws + 25*4096:
//   ibase + 0   : counts[25]
//   ibase + 32  : tileStart[26]   (tileStart[25] == totalTiles)
//   ibase + 64  : bucketEnd[25]
//   ibase + 96  : cursors[25]
//   ibase + 128 : rowIdx[NPAD]
// ---------------------------------------------------------------------------

__global__ void rgcn_zero(int* __restrict__ counts, int* __restrict__ rowIdx) {
    int i = blockIdx.x * blockDim.x + threadIdx.x;
    if (i < NPAD) rowIdx[i] = 0;          // padding rows gather row 0 (discarded)
    if (i < NREL) counts[i] = 0;
}

// Wpack = basis-mix of kernel, written directly in WMMA-B lane layout
__global__ void rgcn_wmat(const float* __restrict__ lin,
                          const float* __restrict__ kern,
                          float* __restrict__ Wpack) {
    int i = blockIdx.x * blockDim.x + threadIdx.x;   // < 25*4096
    if (i >= NREL * WSZ) return;
    int r     = i >> 12;
    int rest  = i & 4095;
    int k     = rest >> 8;
    int rest2 = rest & 255;
    int g     = rest2 >> 7;
    int j     = rest2 & 127;
    int lane  = j >> 2;
    int q     = j & 3;
    int hi    = lane >> 4;
    int nlo   = lane & 15;
    int f     = 4 * k + 2 * hi + (q & 1);
    int o     = (2 * g + (q >> 1)) * 16 + nlo;
    float acc = 0.f;
#pragma unroll
    for (int b = 0; b < NBASE; ++b)
        acc += lin[r * NBASE + b] * kern[b * WSZ + f * UNITS + o];
    Wpack[i] = acc;
}

__global__ void rgcn_hist(const int* __restrict__ rel, int* __restrict__ counts) {
    __shared__ int lc[NREL];
    int tid = threadIdx.x;
    if (tid < NREL) lc[tid] = 0;
    __syncthreads();
    int n = blockIdx.x * blockDim.x + tid;
    if (n < NROWS) atomicAdd(&lc[rel[n]], 1);
    __syncthreads();
    if (tid < NREL && lc[tid] > 0) atomicAdd(&counts[tid], lc[tid]);
}

__global__ void rgcn_scan(const int* __restrict__ counts,
                          int* __restrict__ tileStart,
                          int* __restrict__ bucketEnd,
                          int* __restrict__ cursors) {
    if (threadIdx.x != 0 || blockIdx.x != 0) return;
    int run = 0;
    for (int r = 0; r < NREL; ++r) {
        tileStart[r] = run;
        int c  = counts[r];
        int ps = run << 4;
        cursors[r]   = ps;
        bucketEnd[r] = ps + c;
        run += (c + 15) >> 4;
    }
    tileStart[NREL] = run;
}

__global__ void rgcn_scatter(const int* __restrict__ rel,
                             int* __restrict__ cursors,
                             int* __restrict__ rowIdx) {
    __shared__ int lcount[NREL];
    __shared__ int lbase[NREL];
    int tid = threadIdx.x;
    if (tid < NREL) lcount[tid] = 0;
    __syncthreads();
    int n = blockIdx.x * blockDim.x + tid;
    int r = 0, local = 0;
    if (n < NROWS) {
        r = rel[n];
        local = atomicAdd(&lcount[r], 1);
    }
    __syncthreads();
    if (tid < NREL && lcount[tid] > 0)
        lbase[tid] = atomicAdd(&cursors[tid], lcount[tid]);
    __syncthreads();
    if (n < NROWS) rowIdx[lbase[r] + local] = n;
}

// ---------------------------------------------------------------------------
// Grouped GEMM, one wave32 per 16-row tile, 8 waves (8 tiles) per block.
// All tile/bucket control state is forced into SGPRs via readfirstlane so the
// LDS-vs-global B branch is a scalar branch and EXEC is all-ones at every WMMA.
// ---------------------------------------------------------------------------
__global__ void __launch_bounds__(256)
rgcn_gemm(const float* __restrict__ feat,
          const float* __restrict__ Wpack,
          const float* __restrict__ bias,
          const int*   __restrict__ tileStart,
          const int*   __restrict__ bucketEnd,
          const int*   __restrict__ rowIdx,
          float*       __restrict__ out) {
    __shared__ float ldsW[WSZ];            // 16 KB

    const int lane   = threadIdx.x & 31;
    const int waveId = __builtin_amdgcn_readfirstlane((int)(threadIdx.x >> 5));
    const int tile   = blockIdx.x * 8 + waveId;          // SGPR
    const int total  = __builtin_amdgcn_readfirstlane(tileStart[NREL]);

    // bucket of the block's first tile (clamped) — staged into LDS by everyone
    const int t0 = min((int)(blockIdx.x * 8), total - 1);
    int rFirst = 0;
#pragma unroll
    for (int i = 1; i < NREL; ++i)
        if (t0 >= __builtin_amdgcn_readfirstlane(tileStart[i])) rFirst = i;

    {
        const v4f* src = (const v4f*)(Wpack + (long)rFirst * WSZ);
        v4f*       dst = (v4f*)ldsW;
#pragma unroll
        for (int t = 0; t < 4; ++t)
            dst[threadIdx.x + t * 256] = src[threadIdx.x + t * 256];
    }
    __syncthreads();

    if (tile >= total) return;             // scalar branch; after the barrier

    int r = 0;
#pragma unroll
    for (int i = 1; i < NREL; ++i)
        if (tile >= __builtin_amdgcn_readfirstlane(tileStart[i])) r = i;

    const int posBase = tile << 4;         // SGPR
    const int hi  = lane >> 4;
    const int nlo = lane & 15;

    // gather A (32-bit 16x4 layout): my row is m = nlo
    const int   myRow = rowIdx[posBase + nlo];
    const float* frow = feat + (long)myRow * FDIM;
    v2f a[16];
#pragma unroll
    for (int k = 0; k < 16; ++k)
        a[k] = *(const v2f*)(frow + k * 4 + hi * 2);

    v8f c0 = {}, c1 = {}, c2 = {}, c3 = {};

#define GEMM_STEP(Q0, Q1)                                                                              \
    {                                                                                                  \
        v2f b;                                                                                         \
        b.x = (Q0).x; b.y = (Q0).y;                                                                    \
        c0 = __builtin_amdgcn_wmma_f32_16x16x4_f32(false, a[k], false, b, (short)0, c0, false, false); \
        b.x = (Q0).z; b.y = (Q0).w;                                                                    \
        c1 = __builtin_amdgcn_wmma_f32_16x16x4_f32(false, a[k], false, b, (short)0, c1, false, false); \
        b.x = (Q1).x; b.y = (Q1).y;                                                                    \
        c2 = __builtin_amdgcn_wmma_f32_16x16x4_f32(false, a[k], false, b, (short)0, c2, false, false); \
        b.x = (Q1).z; b.y = (Q1).w;                                                                    \
        c3 = __builtin_amdgcn_wmma_f32_16x16x4_f32(false, a[k], false, b, (short)0, c3, false, false); \
    }

    if (r == rFirst) {                     // scalar (SCC) branch: B from LDS
#pragma unroll
        for (int k = 0; k < 16; ++k) {
            v4f q0 = *(const v4f*)(ldsW + (k * 2 + 0) * 128 + lane * 4);
            v4f q1 = *(const v4f*)(ldsW + (k * 2 + 1) * 128 + lane * 4);
            GEMM_STEP(q0, q1)
        }
    } else {                               // bucket-boundary waves: packed global B
        const float* gw = Wpack + (long)r * WSZ;
#pragma unroll
        for (int k = 0; k < 16; ++k) {
            v4f q0 = *(const v4f*)(gw + (k * 2 + 0) * 128 + lane * 4);
            v4f q1 = *(const v4f*)(gw + (k * 2 + 1) * 128 + lane * 4);
            GEMM_STEP(q0, q1)
        }
    }
#undef GEMM_STEP

    const float b0 = bias[nlo];
    const float b1 = bias[16 + nlo];
    const float b2 = bias[32 + nlo];
    const float b3 = bias[48 + nlo];
    const int endPos = __builtin_amdgcn_readfirstlane(bucketEnd[r]);

#pragma unroll
    for (int j = 0; j < 8; ++j) {
        int p = posBase + j + hi * 8;      // padded position of row M = j + 8*hi
        if (p < endPos) {                  // mask padding rows (stores only)
            int row = rowIdx[p];
            float* orow = out + (long)row * UNITS;
            orow[nlo]      = c0[j] + b0;
            orow[16 + nlo] = c1[j] + b1;
            orow[32 + nlo] = c2[j] + b2;
            orow[48 + nlo] = c3[j] + b3;
        }
    }
}

extern "C" void kernel_launch(void* const* d_in, const int* in_sizes, int n_in,
                              void* d_out, int out_size, void* d_ws, size_t ws_size,
                              hipStream_t stream) {
    const float* feat = (const float*)d_in[0];   // [65536, 64]
    const int*   rel  = (const int*)  d_in[1];   // [65536]
    const float* kern = (const float*)d_in[2];   // [8, 64, 64]
    const float* lin  = (const float*)d_in[3];   // [25, 8]
    const float* bias = (const float*)d_in[4];   // [64]
    float*       out  = (float*)d_out;

    float* Wpack     = (float*)d_ws;
    int*   ibase     = (int*)d_ws + NREL * WSZ;
    int*   counts    = ibase;
    int*   tileStart = ibase + 32;
    int*   bucketEnd = ibase + 64;
    int*   cursors   = ibase + 96;
    int*   rowIdx    = ibase + 128;

    rgcn_zero   <<<(NPAD + 255) / 256, 256, 0, stream>>>(counts, rowIdx);
    rgcn_wmat   <<<(NREL * WSZ + 255) / 256, 256, 0, stream>>>(lin, kern, Wpack);
    rgcn_hist   <<<NROWS / 256, 256, 0, stream>>>(rel, counts);
    rgcn_scan   <<<1, 32, 0, stream>>>(counts, tileStart, bucketEnd, cursors);
    rgcn_scatter<<<NROWS / 256, 256, 0, stream>>>(rel, cursors, rowIdx);

    int blocks = (MAXTILES + 7) / 8;       // 8 waves = 8 tiles per block
    rgcn_gemm<<<blocks, 256, 0, stream>>>(feat, Wpack, bias, tileStart, bucketEnd, rowIdx, out);
}